// GATLayerWithSkip_57191784514100
// MI455X (gfx1250) — compile-verified
//
#include <hip/hip_runtime.h>
#include <hip/hip_bf16.h>
#include <math.h>

#define NNODES   50000
#define INDIM    256
#define OUTDIM   64
#define HEADS    4
#define NEDGES   800000
#define ETOT     (NEDGES + NNODES)   // with self loops
#define NEGSLOPE 0.2f

typedef __attribute__((ext_vector_type(2))) float v2f;
typedef __attribute__((ext_vector_type(4))) float v4f;
typedef __attribute__((ext_vector_type(8))) float v8f;

// ---------- order-preserving float <-> uint for atomic max ----------
__device__ __forceinline__ unsigned f2ord(float f) {
    unsigned u = __float_as_uint(f);
    return (u & 0x80000000u) ? ~u : (u | 0x80000000u);
}
__device__ __forceinline__ float ord2f(unsigned u) {
    return (u & 0x80000000u) ? __uint_as_float(u & 0x7fffffffu)
                             : __uint_as_float(~u);
}

// ---------------- K0: zero output + softmax state ----------------
__global__ void k0_init(float* __restrict__ out, unsigned* __restrict__ mxord,
                        float* __restrict__ den) {
    int i = blockIdx.x * blockDim.x + threadIdx.x;
    if (i < NNODES * OUTDIM) out[i] = 0.0f;
    if (i < NNODES * HEADS) { mxord[i] = 0u; den[i] = 0.0f; }
}

// ---------------- K1: fused WMMA f32 GEMM:  xw = x@W ; res = x@res_w -------
// One wave per 16x16 output tile over the concatenated [256|64] column space.
// Exact f32 via V_WMMA_F32_16X16X4_F32.  A/B fragment layout: K = g + 2*(lane/16).
__global__ void k1_gemm(const float* __restrict__ x, const float* __restrict__ W,
                        const float* __restrict__ res_w,
                        float* __restrict__ xw, float* __restrict__ res) {
    const int NCOLT = (INDIM + OUTDIM) / 16;     // 20 col tiles
    const int NROWT = NNODES / 16;               // 3125 row tiles (exact)
    const int lane  = threadIdx.x & 31;
    const int wid   = blockIdx.x * 8 + (threadIdx.x >> 5);
    if (wid >= NROWT * NCOLT) return;

    const int rowt = wid / NCOLT;
    const int colt = wid % NCOLT;
    const int m0   = rowt * 16;
    const int n0   = colt * 16;
    const int half = lane >> 4;      // 0: K slots {0,1}; 1: K slots {2,3}
    const int l16  = lane & 15;

    const float* __restrict__ B;
    int ldb, bn;
    if (n0 < INDIM) { B = W;     ldb = HEADS * OUTDIM; bn = n0 + l16;         }
    else            { B = res_w; ldb = OUTDIM;         bn = n0 - INDIM + l16; }

    const float* __restrict__ arow = x + (size_t)(m0 + l16) * INDIM;

    v8f c = {};
    for (int kk = 0; kk < INDIM; kk += 4) {
        const int kb = kk + 2 * half;                 // even -> 8B aligned
        v2f a = *(const v2f*)(arow + kb);             // A[m, kb], A[m, kb+1]
        v2f b;
        b.x = B[(size_t)(kb    ) * ldb + bn];         // B[kb,   n]
        b.y = B[(size_t)(kb + 1) * ldb + bn];         // B[kb+1, n]
        c = __builtin_amdgcn_wmma_f32_16x16x4_f32(
                false, a, false, b, (short)0, c, false, false);
    }

    // C/D layout: VGPR v, lane l -> M = v + 8*(l/16), N = l%16
    if (n0 < INDIM) {
        float* dst = xw + (size_t)(m0 + 8 * half) * INDIM + n0 + l16;
        #pragma unroll
        for (int v = 0; v < 8; ++v) dst[(size_t)v * INDIM] = c[v];
    } else {
        float* dst = res + (size_t)(m0 + 8 * half) * OUTDIM + (n0 - INDIM) + l16;
        #pragma unroll
        for (int v = 0; v < 8; ++v) dst[(size_t)v * OUTDIM] = c[v];
    }
}

// ---------------- K2: per-node attention halves (wave per node) -------------
__global__ void k2_attn(const float* __restrict__ xw,
                        const float* __restrict__ att_src,
                        const float* __restrict__ att_dst,
                        float* __restrict__ a_src, float* __restrict__ a_dst) {
    const int lane = threadIdx.x & 31;
    const int node = blockIdx.x * 8 + (threadIdx.x >> 5);
    if (node >= NNODES) return;
    const float* row = xw + (size_t)node * (HEADS * OUTDIM);
    const int c0 = lane * 8;          // 8 channels per lane; head = lane/8
    const int h  = lane >> 3;

    v4f x0 = *(const v4f*)(row + c0);
    v4f x1 = *(const v4f*)(row + c0 + 4);
    v4f s0 = *(const v4f*)(att_src + c0);
    v4f s1 = *(const v4f*)(att_src + c0 + 4);
    v4f d0 = *(const v4f*)(att_dst + c0);
    v4f d1 = *(const v4f*)(att_dst + c0 + 4);

    float ps = x0.x*s0.x + x0.y*s0.y + x0.z*s0.z + x0.w*s0.w
             + x1.x*s1.x + x1.y*s1.y + x1.z*s1.z + x1.w*s1.w;
    float pd = x0.x*d0.x + x0.y*d0.y + x0.z*d0.z + x0.w*d0.w
             + x1.x*d1.x + x1.y*d1.y + x1.z*d1.z + x1.w*d1.w;

    #pragma unroll
    for (int off = 1; off < 8; off <<= 1) {
        ps += __shfl_xor(ps, off, 32);
        pd += __shfl_xor(pd, off, 32);
    }
    if ((lane & 7) == 0) {
        a_src[node * HEADS + h] = ps;
        a_dst[node * HEADS + h] = pd;
    }
}

__device__ __forceinline__ void edge_nodes(const int* __restrict__ ei, int e,
                                           int& s, int& d) {
    if (e < NEDGES) { s = ei[e]; d = ei[NEDGES + e]; }
    else            { s = d = e - NEDGES; }           // self loop
}

__device__ __forceinline__ float lrelu(float v) {
    return v > 0.0f ? v : NEGSLOPE * v;
}

// ---------------- K3: segment max of leaky-relu logits ----------------------
__global__ void k3_max(const int* __restrict__ ei, const float* __restrict__ a_src,
                       const float* __restrict__ a_dst, unsigned* __restrict__ mxord) {
    int e = blockIdx.x * blockDim.x + threadIdx.x;
    if (e >= ETOT) return;
    int s, d; edge_nodes(ei, e, s, d);
    v4f as = *(const v4f*)(a_src + (size_t)s * HEADS);
    v4f ad = *(const v4f*)(a_dst + (size_t)d * HEADS);
    unsigned* m = mxord + (size_t)d * HEADS;
    atomicMax(m + 0, f2ord(lrelu(as.x + ad.x)));
    atomicMax(m + 1, f2ord(lrelu(as.y + ad.y)));
    atomicMax(m + 2, f2ord(lrelu(as.z + ad.z)));
    atomicMax(m + 3, f2ord(lrelu(as.w + ad.w)));
}

// ---------------- K4: exp(logit - max), denominator sums --------------------
__global__ void k4_expsum(const int* __restrict__ ei, const float* __restrict__ a_src,
                          const float* __restrict__ a_dst,
                          const unsigned* __restrict__ mxord,
                          float* __restrict__ den, float* __restrict__ wbuf) {
    int e = blockIdx.x * blockDim.x + threadIdx.x;
    if (e >= ETOT) return;
    int s, d; edge_nodes(ei, e, s, d);
    v4f as = *(const v4f*)(a_src + (size_t)s * HEADS);
    v4f ad = *(const v4f*)(a_dst + (size_t)d * HEADS);
    const unsigned* m = mxord + (size_t)d * HEADS;
    v4f w;
    w.x = expf(lrelu(as.x + ad.x) - ord2f(m[0]));
    w.y = expf(lrelu(as.y + ad.y) - ord2f(m[1]));
    w.z = expf(lrelu(as.z + ad.z) - ord2f(m[2]));
    w.w = expf(lrelu(as.w + ad.w) - ord2f(m[3]));
    *(v4f*)(wbuf + (size_t)e * HEADS) = w;
    float* dn = den + (size_t)d * HEADS;
    atomicAdd(dn + 0, w.x);
    atomicAdd(dn + 1, w.y);
    atomicAdd(dn + 2, w.z);
    atomicAdd(dn + 3, w.w);
}

// ---------------- K5: weighted scatter, head-mean folded in -----------------
// One wave per edge; lane covers channels {lane, lane+32}; 64 atomics/edge.
__global__ void k5_scatter(const int* __restrict__ ei, const float* __restrict__ wbuf,
                           const float* __restrict__ den, const float* __restrict__ xw,
                           float* __restrict__ out) {
    const int lane = threadIdx.x & 31;
    const int e    = blockIdx.x * 8 + (threadIdx.x >> 5);
    if (e >= ETOT) return;
    int s, d; edge_nodes(ei, e, s, d);

    v4f w  = *(const v4f*)(wbuf + (size_t)e * HEADS);
    v4f dn = *(const v4f*)(den + (size_t)d * HEADS);
    const float a0 = 0.25f * w.x / (dn.x + 1e-16f);   // 0.25 = mean over heads
    const float a1 = 0.25f * w.y / (dn.y + 1e-16f);
    const float a2 = 0.25f * w.z / (dn.z + 1e-16f);
    const float a3 = 0.25f * w.w / (dn.w + 1e-16f);

    const float* sr = xw + (size_t)s * (HEADS * OUTDIM);
    float v0 = a0 * sr[lane      ] + a1 * sr[64  + lane] +
               a2 * sr[128 + lane] + a3 * sr[192 + lane];
    float v1 = a0 * sr[32  + lane] + a1 * sr[96  + lane] +
               a2 * sr[160 + lane] + a3 * sr[224 + lane];
    float* o = out + (size_t)d * OUTDIM;
    atomicAdd(o + lane,      v0);
    atomicAdd(o + 32 + lane, v1);
}

// ---------------- K6: + residual + bias + res_b -----------------------------
__global__ void k6_final(float* __restrict__ out, const float* __restrict__ res,
                         const float* __restrict__ bias, const float* __restrict__ res_b) {
    int i = blockIdx.x * blockDim.x + threadIdx.x;
    if (i >= NNODES * OUTDIM) return;
    int c = i & (OUTDIM - 1);
    out[i] += res[i] + bias[c] + res_b[c];
}

extern "C" void kernel_launch(void* const* d_in, const int* in_sizes, int n_in,
                              void* d_out, int out_size, void* d_ws, size_t ws_size,
                              hipStream_t stream) {
    const float* x       = (const float*)d_in[0];
    const int*   ei      = (const int*)  d_in[1];   // [2, E] int32
    const float* W       = (const float*)d_in[2];
    const float* att_src = (const float*)d_in[3];
    const float* att_dst = (const float*)d_in[4];
    const float* bias    = (const float*)d_in[5];
    const float* res_w   = (const float*)d_in[6];
    const float* res_b   = (const float*)d_in[7];
    float* out = (float*)d_out;

    // scratch layout (all offsets multiples of 256B)
    char* ws = (char*)d_ws;
    float*    xw    = (float*)   (ws);                                   // N*256 f32
    float*    res   = (float*)   (ws + (size_t)NNODES * INDIM * 4);      // N*64
    float*    a_src = (float*)   (ws + (size_t)NNODES * (INDIM + OUTDIM) * 4);
    float*    a_dst = a_src + (size_t)NNODES * HEADS;
    unsigned* mxord = (unsigned*)(a_dst + (size_t)NNODES * HEADS);
    float*    den   = (float*)   (mxord + (size_t)NNODES * HEADS);
    float*    wbuf  = den + (size_t)NNODES * HEADS;                      // Et*4

    // K0: init
    {
        int total = NNODES * OUTDIM;   // >= NNODES*HEADS
        k0_init<<<(total + 255) / 256, 256, 0, stream>>>(out, mxord, den);
    }
    // K1: WMMA GEMM (3125*20 = 62500 tiles, 8 waves/block)
    {
        int tiles = (NNODES / 16) * ((INDIM + OUTDIM) / 16);
        k1_gemm<<<(tiles + 7) / 8, 256, 0, stream>>>(x, W, res_w, xw, res);
    }
    // K2: attention halves (wave per node)
    k2_attn<<<(NNODES + 7) / 8, 256, 0, stream>>>(xw, att_src, att_dst, a_src, a_dst);
    // K3: segment max
    k3_max<<<(ETOT + 255) / 256, 256, 0, stream>>>(ei, a_src, a_dst, mxord);
    // K4: exp + denominator
    k4_expsum<<<(ETOT + 255) / 256, 256, 0, stream>>>(ei, a_src, a_dst, mxord, den, wbuf);
    // K5: scatter aggregation (wave per edge)
    k5_scatter<<<(ETOT + 7) / 8, 256, 0, stream>>>(ei, wbuf, den, xw, out);
    // K6: epilogue
    k6_final<<<(NNODES * OUTDIM + 255) / 256, 256, 0, stream>>>(out, res, bias, res_b);
}